// DistributionLoss_42812234006951
// MI455X (gfx1250) — compile-verified
//
#include <hip/hip_runtime.h>
#include <cstdint>
#include <cstddef>

// Problem geometry (from reference): B=2, C=8, D=H=W=128
#define NCH 16                 // B*C channels
#define SPAT 2097152           // 128^3 elements per channel
#define BPC 128                // blocks per channel
#define WAVES 8                // waves per block (256 threads, wave32)
#define STAGE_ELEMS 128        // 32 lanes * float4 per async stage
#define EPW (SPAT / (BPC * WAVES))      // 2048 elements per wave
#define STAGES (EPW / STAGE_ELEMS)      // 16 pipeline stages

static_assert(SPAT % (BPC * WAVES * STAGE_ELEMS) == 0, "exact tiling required");

// ---- CDNA5 async-to-LDS pipeline primitives (gfx1250) ----
// GVS mode: lds_byte_addr (VGPR), 32-bit per-lane byte offset (VGPR), 64-bit base (SGPR pair)
#define ASYNC_B128(ldsoff, voff, base)                                  \
  asm volatile("global_load_async_to_lds_b128 %0, %1, %2"               \
               :: "v"(ldsoff), "v"(voff), "s"(base) : "memory")
#define WAIT_ASYNCCNT(n)                                                \
  asm volatile("s_wait_asynccnt " #n ::: "memory")
#define WAIT_DSCNT0()                                                   \
  asm volatile("s_wait_dscnt 0x0" ::: "memory")

__device__ __forceinline__ unsigned lds_addr_of(const void* p) {
  // addrspacecast(LDS->flat) puts the wave-relative LDS byte offset in the low 32 bits
  return (unsigned)(size_t)p;
}

__device__ __forceinline__ float clampN(float v) {   // normalize-side clamp
  return fminf(fmaxf(v, 1e-6f), 1e6f);
}
__device__ __forceinline__ float clampP(float v) {   // KL-side clamp
  return fminf(fmaxf(v, 1e-24f), 1.0f - 1e-24f);
}

__device__ __forceinline__ double wave_reduce(double v) {
  #pragma unroll
  for (int o = 16; o > 0; o >>= 1) v += __shfl_down(v, o, 32);
  return v;
}

// Reduce one double per thread to a single value on (wave0, lane0).
__device__ __forceinline__ double block_reduce(double v, double* scratch,
                                               int lane, int wave) {
  v = wave_reduce(v);
  if (lane == 0) scratch[wave] = v;
  __syncthreads();
  double r = 0.0;
  if (wave == 0) {
    r = (lane < WAVES) ? scratch[lane] : 0.0;
    #pragma unroll
    for (int o = 4; o > 0; o >>= 1) r += __shfl_down(r, o, 32);
  }
  __syncthreads();   // allow scratch reuse
  return r;
}

// ---------------- Pass 1: per-(channel, block) partial sums ----------------
__global__ __launch_bounds__(256)
void js_pass1(const float* __restrict__ x, const float* __restrict__ y,
              double* __restrict__ psx, double* __restrict__ psy) {
  __shared__ float4 xb[2][WAVES][32];
  __shared__ float4 yb[2][WAVES][32];
  __shared__ double red[WAVES];

  const int tid  = threadIdx.x;
  const int lane = tid & 31;
  const int wave = tid >> 5;
  const int ch   = blockIdx.y;
  const int blk  = blockIdx.x;

  const unsigned wave_elem = (unsigned)ch * (unsigned)SPAT
                           + (unsigned)(blk * WAVES + wave) * (unsigned)EPW;
  const unsigned voff0 = wave_elem * 4u + (unsigned)lane * 16u;

  const unsigned lx0 = lds_addr_of(&xb[0][wave][lane]);
  const unsigned lx1 = lds_addr_of(&xb[1][wave][lane]);
  const unsigned ly0 = lds_addr_of(&yb[0][wave][lane]);
  const unsigned ly1 = lds_addr_of(&yb[1][wave][lane]);

  // Prologue: stage 0 in flight
  ASYNC_B128(lx0, voff0, x);
  ASYNC_B128(ly0, voff0, y);

  double sx = 0.0, sy = 0.0;
  #pragma unroll 2
  for (int i = 0; i < STAGES; ++i) {
    const int buf = i & 1;
    if (i + 1 < STAGES) {
      // Ensure iteration i-1's DS reads of buf^1 have retired before the
      // async engine is allowed to overwrite that buffer (WAR safety; the
      // reads completed long ago, this is effectively free).
      WAIT_DSCNT0();
      const unsigned voff = voff0 + (unsigned)((i + 1) * STAGE_ELEMS * 4);
      ASYNC_B128(buf ? lx0 : lx1, voff, x);
      ASYNC_B128(buf ? ly0 : ly1, voff, y);
      WAIT_ASYNCCNT(2);   // previous stage's 2 ops have landed in LDS
    } else {
      WAIT_ASYNCCNT(0);   // drain
    }
    const float4 xv = xb[buf][wave][lane];
    const float4 yv = yb[buf][wave][lane];
    // Stage-local f32 partial (2-ulp class error), cross-stage f64 accumulate:
    // cuts f64 cvt/add VALU pressure 4x vs per-element f64.
    const float fx = (clampN(xv.x) + clampN(xv.y))
                   + (clampN(xv.z) + clampN(xv.w));
    const float fy = (clampN(yv.x) + clampN(yv.y))
                   + (clampN(yv.z) + clampN(yv.w));
    sx += (double)fx;
    sy += (double)fy;
  }

  const double bx = block_reduce(sx, red, lane, wave);
  const double by = block_reduce(sy, red, lane, wave);
  if (tid == 0) {
    psx[ch * BPC + blk] = bx;
    psy[ch * BPC + blk] = by;
  }
}

// ---------------- Mid: fold block partials into per-channel sums ----------------
__global__ void js_mid(const double* __restrict__ psx,
                       const double* __restrict__ psy,
                       double* __restrict__ sums) {
  const int c = threadIdx.x;
  if (c < NCH) {
    double s = 0.0, t = 0.0;
    for (int b = 0; b < BPC; ++b) {       // fixed order -> deterministic
      s += psx[c * BPC + b];
      t += psy[c * BPC + b];
    }
    sums[c]       = fmax(s, 1e-24);
    sums[NCH + c] = fmax(t, 1e-24);
  }
}

// ---------------- Pass 2: KL(p||m) and KL(q||m) partials ----------------
__global__ __launch_bounds__(256)
void js_pass2(const float* __restrict__ x, const float* __restrict__ y,
              const double* __restrict__ sums,
              double* __restrict__ pkP, double* __restrict__ pkQ) {
  __shared__ float4 xb[2][WAVES][32];
  __shared__ float4 yb[2][WAVES][32];
  __shared__ double red[WAVES];

  const int tid  = threadIdx.x;
  const int lane = tid & 31;
  const int wave = tid >> 5;
  const int ch   = blockIdx.y;
  const int blk  = blockIdx.x;

  const float invSx = (float)(1.0 / sums[ch]);
  const float invSy = (float)(1.0 / sums[NCH + ch]);

  const unsigned wave_elem = (unsigned)ch * (unsigned)SPAT
                           + (unsigned)(blk * WAVES + wave) * (unsigned)EPW;
  const unsigned voff0 = wave_elem * 4u + (unsigned)lane * 16u;

  const unsigned lx0 = lds_addr_of(&xb[0][wave][lane]);
  const unsigned lx1 = lds_addr_of(&xb[1][wave][lane]);
  const unsigned ly0 = lds_addr_of(&yb[0][wave][lane]);
  const unsigned ly1 = lds_addr_of(&yb[1][wave][lane]);

  ASYNC_B128(lx0, voff0, x);
  ASYNC_B128(ly0, voff0, y);

  double klP = 0.0, klQ = 0.0;
  #pragma unroll 2
  for (int i = 0; i < STAGES; ++i) {
    const int buf = i & 1;
    if (i + 1 < STAGES) {
      WAIT_DSCNT0();      // WAR safety before overwriting buf^1
      const unsigned voff = voff0 + (unsigned)((i + 1) * STAGE_ELEMS * 4);
      ASYNC_B128(buf ? lx0 : lx1, voff, x);
      ASYNC_B128(buf ? ly0 : ly1, voff, y);
      WAIT_ASYNCCNT(2);
    } else {
      WAIT_ASYNCCNT(0);
    }
    const float4 xv = xb[buf][wave][lane];
    const float4 yv = yb[buf][wave][lane];

    float tP = 0.0f, tQ = 0.0f;
    const float px[4] = {xv.x, xv.y, xv.z, xv.w};
    const float qx[4] = {yv.x, yv.y, yv.z, yv.w};
    #pragma unroll
    for (int k = 0; k < 4; ++k) {
      const float p  = clampN(px[k]) * invSx;
      const float q  = clampN(qx[k]) * invSy;
      const float pc = clampP(p);
      const float qc = clampP(q);
      const float mc = clampP(0.5f * (p + q));
      const float lm = __logf(mc);            // v_log_f32 path
      tP += pc * (__logf(pc) - lm);
      tQ += qc * (__logf(qc) - lm);
    }
    klP += (double)tP;
    klQ += (double)tQ;
  }

  const double bP = block_reduce(klP, red, lane, wave);
  const double bQ = block_reduce(klQ, red, lane, wave);
  if (tid == 0) {
    pkP[ch * BPC + blk] = bP;
    pkQ[ch * BPC + blk] = bQ;
  }
}

// ---------------- Final: clip per-channel KLs, JS, mean ----------------
__global__ void js_final(const double* __restrict__ pkP,
                         const double* __restrict__ pkQ,
                         float* __restrict__ out) {
  __shared__ double js[NCH];
  const int c = threadIdx.x;
  if (c < NCH) {
    double a = 0.0, b = 0.0;
    for (int i = 0; i < BPC; ++i) {
      a += pkP[c * BPC + i];
      b += pkQ[c * BPC + i];
    }
    a = fmin(fmax(a, 0.0), 1000.0);
    b = fmin(fmax(b, 0.0), 1000.0);
    js[c] = 0.5 * (a + b);
  }
  __syncthreads();
  if (c == 0) {
    double s = 0.0;
    for (int i = 0; i < NCH; ++i) s += js[i];   // fixed order
    out[0] = (float)(s / (double)NCH);
  }
}

extern "C" void kernel_launch(void* const* d_in, const int* in_sizes, int n_in,
                              void* d_out, int out_size, void* d_ws, size_t ws_size,
                              hipStream_t stream) {
  const float* x = (const float*)d_in[0];   // heatmaps, fp32, 16*2^21 elems
  const float* y = (const float*)d_in[1];   // gt
  (void)in_sizes; (void)n_in; (void)out_size; (void)ws_size;

  double* ws   = (double*)d_ws;
  double* psx  = ws;                    // [NCH*BPC]
  double* psy  = psx + NCH * BPC;       // [NCH*BPC]
  double* pkP  = psy + NCH * BPC;       // [NCH*BPC]
  double* pkQ  = pkP + NCH * BPC;       // [NCH*BPC]
  double* sums = pkQ + NCH * BPC;       // [2*NCH]  (Sx then Sy)

  dim3 grid(BPC, NCH);
  js_pass1<<<grid, 256, 0, stream>>>(x, y, psx, psy);
  js_mid  <<<1, 32, 0, stream>>>(psx, psy, sums);
  js_pass2<<<grid, 256, 0, stream>>>(x, y, sums, pkP, pkQ);
  js_final<<<1, 32, 0, stream>>>(pkP, pkQ, (float*)d_out);
}